// TensorSnake_72000831750192
// MI455X (gfx1250) — compile-verified
//
#include <hip/hip_runtime.h>
#include <stdint.h>

// ---------------------------------------------------------------------------
// TensorSnake step for MI455X (gfx1250, wave32).
// One game (32x32 board) per wave: lane = row, 32 regs = columns.
// Board staged global -> LDS -> global with gfx1250 async LDS ops (ASYNCcnt).
// Uniform-index board accesses (probe / food / head) done as single DS ops.
// ---------------------------------------------------------------------------

#define WAVES_PER_BLOCK 8
#define THREADS_PER_BLOCK (WAVES_PER_BLOCK * 32)

#if defined(__HIP_DEVICE_COMPILE__) && \
    __has_builtin(__builtin_amdgcn_global_load_async_to_lds_b128) && \
    __has_builtin(__builtin_amdgcn_global_store_async_from_lds_b128)
#define HAVE_ASYNC_LDS 1
#else
#define HAVE_ASYNC_LDS 0
#endif

typedef int v4i __attribute__((vector_size(16)));
typedef v4i __attribute__((address_space(1))) g_v4i;   // global (AS1)
typedef v4i __attribute__((address_space(3))) l_v4i;   // LDS    (AS3)

__device__ __forceinline__ g_v4i* as_gbl(const void* p) {
    return (g_v4i*)(unsigned long long)p;
}
__device__ __forceinline__ l_v4i* as_lds(void* p) {
    // Flat LDS address: addr[31:0] == LDS byte offset (aperture layout),
    // so truncation yields a valid AS(3) pointer value.
    return (l_v4i*)(unsigned int)(unsigned long long)p;
}

__device__ __forceinline__ void wait_async0() {
#if defined(__HIP_DEVICE_COMPILE__)
#if __has_builtin(__builtin_amdgcn_s_wait_asynccnt)
    __builtin_amdgcn_s_wait_asynccnt(0);
#else
    asm volatile("s_wait_asynccnt 0" ::: "memory");
#endif
#endif
}

__global__ void __launch_bounds__(THREADS_PER_BLOCK)
tensor_snake_step(const float* __restrict__ state,   // [G,32,32]
                  const int*   __restrict__ pos_prev,// [G,2]
                  const int*   __restrict__ pos_cur, // [G,2]
                  const int*   __restrict__ action,  // [G]
                  const float* __restrict__ rand_u,  // [G]
                  float*       __restrict__ out,     // [G,32,32]
                  int G)
{
    __shared__ __align__(16) float lds[WAVES_PER_BLOCK * 1024];

    const int lane = threadIdx.x & 31;
    const int wave = threadIdx.x >> 5;
    const int g    = blockIdx.x * WAVES_PER_BLOCK + wave;
    const int gc   = (g < G) ? g : 0;   // inactive waves shadow game 0 (no store)

    const float* gsrc  = state + (size_t)gc * 1024;
    float*       gdst  = out   + (size_t)gc * 1024;
    float*       lbase = &lds[wave * 1024];

    // ---- Stage board global -> LDS (512B contiguous per instruction) ------
#if HAVE_ASYNC_LDS
#pragma unroll
    for (int i = 0; i < 8; ++i) {
        __builtin_amdgcn_global_load_async_to_lds_b128(
            as_gbl(gsrc + i * 128 + lane * 4),
            as_lds(lbase + i * 128 + lane * 4),
            /*offset=*/0, /*cpol=*/0);
    }
    wait_async0();
#else
#pragma unroll
    for (int i = 0; i < 8; ++i)
        ((float4*)lbase)[i * 32 + lane] = ((const float4*)gsrc)[i * 32 + lane];
#endif
    __syncthreads();

    // ---- Per-game scalars (wave-uniform) -----------------------------------
    const int   ppx = pos_prev[gc * 2 + 0], ppy = pos_prev[gc * 2 + 1];
    const int   pcx = pos_cur [gc * 2 + 0], pcy = pos_cur [gc * 2 + 1];
    const int   act = action[gc];
    const float ru  = rand_u[gc];

    int dx = pcx - ppx, dy = pcy - ppy;
    int ndx = dx, ndy = dy;
    if (act == 0)      { ndx = -dy; ndy =  dx; }   // left
    else if (act == 2) { ndx =  dy; ndy = -dx; }   // right
    int nx = pcx + ndx, ny = pcy + ndy;
    const bool outside = (nx < 0) | (nx >= 32) | (ny < 0) | (ny >= 32);
    const int  cnx = min(max(nx, 0), 31);
    const int  cny = min(max(ny, 0), 31);

    const float cell    = lbase[cnx * 32 + cny];   // original board, uniform addr
    const bool  dead    = outside | (cell > 0.0f);
    const bool  feeding = (cell == -1.0f);
    const bool  spawn   = dead | feeding;

    // ---- Pull this lane's row (row = lane) into registers ------------------
    float r[32];
#pragma unroll
    for (int i = 0; i < 8; ++i) {
        float4 v = ((float4*)lbase)[lane * 8 + i];
        r[i * 4 + 0] = v.x; r[i * 4 + 1] = v.y;
        r[i * 4 + 2] = v.z; r[i * 4 + 3] = v.w;
    }

    int hx = pcx, hy = pcy;   // effective pos_cur
    int tx = cnx, ty = cny;   // effective new-head cell
    if (dead) {               // wave-uniform branch, rare
#pragma unroll
        for (int j = 0; j < 32; ++j) r[j] = 0.0f;
        if (lane == 16) { r[14] = 1.0f; r[15] = 2.0f; r[16] = 1.0f; }
        hx = 16; hy = 15;
        tx = 16; ty = 16;
    }

    // ---- Food index: 1024-cell inclusive cdf of free mask ------------------
    int t = 0;
#pragma unroll
    for (int j = 0; j < 32; ++j) t += (r[j] == 0.0f) ? 1 : 0;

    int x = t;                                   // inclusive scan across lanes
#pragma unroll
    for (int d = 1; d < 32; d <<= 1) {
        int y = __shfl_up(x, d, 32);
        if (lane >= d) x += y;
    }
    const int P     = x - t;                     // exclusive row prefix
    const int total = __shfl(x, 31, 32);
    // float(cum) <= ru*total  <=>  cum <= floor(ru*total)  (cum integer >= 0)
    const int ithr  = (int)floorf(ru * (float)total);

    int cum = P, cnt = 0;
#pragma unroll
    for (int j = 0; j < 32; ++j) {
        cum += (r[j] == 0.0f) ? 1 : 0;
        cnt += (cum <= ithr) ? 1 : 0;
    }
#pragma unroll
    for (int d = 16; d; d >>= 1) cnt += __shfl_xor(cnt, d, 32);
    const int food = min(cnt, 1023);             // flat index into the board

    // ---- Decrement body counters (skip when feeding) -----------------------
    // (Commutes with the food write: dec never touches -1, and a spawned food
    //  cell was free (0), which dec leaves alone.)
    if (!feeding) {           // wave-uniform branch
#pragma unroll
        for (int j = 0; j < 32; ++j) r[j] -= (r[j] > 0.0f) ? 1.0f : 0.0f;
    }

    // ---- Write decremented board regs -> LDS -------------------------------
#pragma unroll
    for (int i = 0; i < 8; ++i) {
        float4 v;
        v.x = r[i * 4 + 0]; v.y = r[i * 4 + 1];
        v.z = r[i * 4 + 2]; v.w = r[i * 4 + 3];
        ((float4*)lbase)[lane * 8 + i] = v;
    }

    // ---- Food + head as single-lane DS ops (same-wave DS is in-order) ------
    if (lane == 0) {
        if (spawn) lbase[food] = -1.0f;
        const float hv = lbase[hx * 32 + hy] + 1.0f;   // post-dec, post-food
        lbase[tx * 32 + ty] = hv;
    }
    __syncthreads();

    // ---- LDS -> global (async, 512B contiguous per instruction) ------------
    if (g < G) {
#if HAVE_ASYNC_LDS
#pragma unroll
        for (int i = 0; i < 8; ++i) {
            __builtin_amdgcn_global_store_async_from_lds_b128(
                as_gbl(gdst + i * 128 + lane * 4),
                as_lds(lbase + i * 128 + lane * 4),
                /*offset=*/0, /*cpol=*/0);
        }
        wait_async0();
#else
#pragma unroll
        for (int i = 0; i < 8; ++i)
            ((float4*)gdst)[i * 32 + lane] = ((float4*)lbase)[i * 32 + lane];
#endif
    }
}

extern "C" void kernel_launch(void* const* d_in, const int* in_sizes, int n_in,
                              void* d_out, int out_size, void* d_ws, size_t ws_size,
                              hipStream_t stream) {
    (void)n_in; (void)out_size; (void)d_ws; (void)ws_size;
    const float* state    = (const float*)d_in[0];
    const int*   pos_prev = (const int*)d_in[1];
    const int*   pos_cur  = (const int*)d_in[2];
    const int*   action   = (const int*)d_in[3];
    const float* rand_u   = (const float*)d_in[4];
    float*       out      = (float*)d_out;

    const int G = in_sizes[0] / 1024;                     // [G,32,32]
    const int blocks = (G + WAVES_PER_BLOCK - 1) / WAVES_PER_BLOCK;
    tensor_snake_step<<<blocks, THREADS_PER_BLOCK, 0, stream>>>(
        state, pos_prev, pos_cur, action, rand_u, out, G);
}